// MemoryAsContext_73040213836404
// MI455X (gfx1250) — compile-verified
//
#include <hip/hip_runtime.h>
#include <math.h>

// ---------------- model constants ----------------
#define DIMV   1024
#define VOCAB  32000
#define HEADS  16
#define HD     64
#define SEGL   128
#define NPT    16
#define WIDTHM 128
#define BATCH  2
#define NTOK   512
#define NSEG   4
#define T_CTX  (NPT + 2*SEGL)          // 272
// per-sample memory-param blob layout (floats)
#define OFF_W1 0
#define OFF_B1 131072                  // 1024*128
#define OFF_W2 131200
#define OFF_B2 262272                  // +128*1024
#define PBLOB  263296

typedef float v2f __attribute__((ext_vector_type(2)));
typedef float v8f __attribute__((ext_vector_type(8)));
typedef int   v4i __attribute__((ext_vector_type(4)));

#if __has_builtin(__builtin_amdgcn_global_load_async_to_lds_b128) && \
    __has_builtin(__builtin_amdgcn_s_wait_asynccnt)
#define HAVE_ASYNC_LDS 1
typedef __attribute__((address_space(1))) v4i as1_v4i;   // global
typedef __attribute__((address_space(3))) v4i as3_v4i;   // LDS
#define ASYNC_CP(srcp, dstp) \
    __builtin_amdgcn_global_load_async_to_lds_b128((as1_v4i*)(srcp), (as3_v4i*)(dstp), 0, 0)
#else
#define HAVE_ASYNC_LDS 0
#define ASYNC_CP(srcp, dstp) (*(float4*)(dstp) = *(const float4*)(srcp))
#endif

__device__ __forceinline__ float siluf(float x) {
    return x / (1.0f + __expf(-x));
}

// ===========================================================================
// Tiled WMMA GEMM for row-major A[M,K], B[K,N] (csA==csB==csC==1):
//   C = act(scale*(A@B) + bias)
// Block = 256 threads (8 waves), C tile 128x128.
// Wave (wm,wn) owns rows wm*64..+64 (4 sub-tiles) x cols wn*32..+32 (2 sub-tiles)
// -> 8 v8f accumulators, A frag reused x2, B frag x4.
// K consumed in chunks of 16 from double-buffered LDS, filled with
// GLOBAL_LOAD_ASYNC_TO_LDS_B128 (ASYNCcnt) when available.
// Requires N%128==0, K%16==0; M ragged rows clamped on load, guarded on store.
// ===========================================================================
#define TM 128
#define TN 128
#define TK 16
#define APAD 20        // A-tile row stride in floats (16B aligned, conflict-free)

__global__ void __launch_bounds__(256)
k_gemm_tiled(const float* __restrict__ A, const float* __restrict__ Bm,
             const float* __restrict__ bias, float* __restrict__ C,
             int M, int N, int K,
             long rsA, long rsB, long rsC,
             long sA, long sB, long sC, long sBias,
             float scale, int act)
{
    __shared__ float As[2][TM][APAD];      // 20 KB
    __shared__ float Bs[2][TK][TN];        // 16 KB

    const int z = blockIdx.z;
    A  += (long)z * sA;
    Bm += (long)z * sB;
    C  += (long)z * sC;

    const int n0  = blockIdx.x * TN;
    const int m0  = blockIdx.y * TM;
    const int tid = threadIdx.x;
    const int wave = tid >> 5;
    const int lane = tid & 31;
    const int wm = wave & 1;               // 0..1
    const int wn = wave >> 1;              // 0..3
    const int half = lane >> 4;
    const int l15  = lane & 15;

    v8f acc[4][2];
#pragma unroll
    for (int i = 0; i < 4; ++i)
#pragma unroll
        for (int j = 0; j < 2; ++j)
            acc[i][j] = (v8f){0.f,0.f,0.f,0.f,0.f,0.f,0.f,0.f};

    // ---- copy one K-chunk (16 wide) of A and B into LDS buffer `buf` ----
    auto copy_chunk = [&](int kc, int buf) {
        // A tile: 128 rows x 16 floats = 512 x 16B; 2 per thread
#pragma unroll
        for (int r = 0; r < 2; ++r) {
            const int idx  = tid + (r << 8);       // 0..511
            const int row  = idx >> 2;
            const int c4   = idx & 3;
            int gm = m0 + row;
            if (gm >= M) gm = M - 1;               // clamp (store is guarded)
            const float* src = A + (long)gm * rsA + kc + (c4 << 2);
            float* dst = &As[buf][row][c4 << 2];
            ASYNC_CP(src, dst);
        }
        // B tile: 16 rows x 128 floats = 512 x 16B; 2 per thread
#pragma unroll
        for (int r = 0; r < 2; ++r) {
            const int idx  = tid + (r << 8);
            const int row  = idx >> 5;
            const int c4   = idx & 31;
            const float* src = Bm + (long)(kc + row) * rsB + n0 + (c4 << 2);
            float* dst = &Bs[buf][row][c4 << 2];
            ASYNC_CP(src, dst);
        }
    };

    auto compute_chunk = [&](int buf) {
#pragma unroll
        for (int kk = 0; kk < TK; kk += 4) {
            const int ka  = kk + (half << 1);
            const int kb0 = kk + half;
            const int kb1 = kk + 2 + half;
            v2f af[4], bf[2];
#pragma unroll
            for (int i = 0; i < 4; ++i) {
                const int mrow = (wm << 6) + (i << 4) + l15;
                af[i] = *(const v2f*)&As[buf][mrow][ka];      // ds_load_b64
            }
#pragma unroll
            for (int j = 0; j < 2; ++j) {
                const int ncol = (wn << 5) + (j << 4) + l15;
                bf[j].x = Bs[buf][kb0][ncol];
                bf[j].y = Bs[buf][kb1][ncol];
            }
#pragma unroll
            for (int i = 0; i < 4; ++i)
#pragma unroll
                for (int j = 0; j < 2; ++j)
                    acc[i][j] = __builtin_amdgcn_wmma_f32_16x16x4_f32(
                        false, af[i], false, bf[j], (short)0, acc[i][j],
                        false, false);
        }
    };

    const int nchunk = K / TK;
    copy_chunk(0, 0);

    for (int c = 0; c < nchunk; ++c) {
        const int buf = c & 1;
        if (c + 1 < nchunk) {
            copy_chunk((c + 1) * TK, buf ^ 1);   // 4 async instrs in flight on top
#if HAVE_ASYNC_LDS
            __builtin_amdgcn_s_wait_asynccnt(4); // retire chunk c (this wave)
#endif
        } else {
#if HAVE_ASYNC_LDS
            __builtin_amdgcn_s_wait_asynccnt(0); // retire everything
#endif
        }
        __syncthreads();
        compute_chunk(buf);
        __syncthreads();
    }

    // ---- epilogue ----
#pragma unroll
    for (int j = 0; j < 2; ++j) {
        const int ncol = n0 + (wn << 5) + (j << 4) + l15;
        float badd = 0.f;
        if (bias) badd = bias[(long)z * sBias + ncol];
#pragma unroll
        for (int i = 0; i < 4; ++i) {
            const int mbase = m0 + (wm << 6) + (i << 4) + (half << 3);
#pragma unroll
            for (int r = 0; r < 8; ++r) {
                const int mrow = mbase + r;
                if (mrow < M) {
                    float v = acc[i][j][r] * scale + badd;
                    if (act == 1) v = siluf(v);
                    C[(long)mrow * rsC + ncol] = v;
                }
            }
        }
    }
}

// ===========================================================================
// Generic strided, batched 1-wave GEMM (for attention: transposed B, N=272).
// ===========================================================================
__global__ void __launch_bounds__(32)
k_gemm(const float* __restrict__ A, const float* __restrict__ Bm,
       const float* __restrict__ bias, float* __restrict__ C,
       int M, int N, int K,
       long rsA, long csA, long rsB, long csB, long rsC,
       int batchInner,
       long sAo, long sAi, long sBo, long sBi, long sCo, long sCi, long sBias,
       float scale, int act)
{
    const int lane = threadIdx.x & 31;
    const int tn   = blockIdx.x << 4;
    const int tm   = blockIdx.y << 4;
    const int z    = blockIdx.z;
    const int bo   = z / batchInner;
    const int bi   = z - bo * batchInner;

    A  += (long)bo * sAo + (long)bi * sAi;
    Bm += (long)bo * sBo + (long)bi * sBi;
    C  += (long)bo * sCo + (long)bi * sCi;

    const int  half = lane >> 4;
    const int  l15  = lane & 15;
    const long aRow = (long)(tm + l15) * rsA;
    const long bCol = (long)(tn + l15) * csB;

    v8f acc = {0.f,0.f,0.f,0.f,0.f,0.f,0.f,0.f};

    for (int k0 = 0; k0 < K; k0 += 4) {
        const int ka = k0 + (half << 1);
        v2f af, bf;
        af.x = A[aRow + (long)ka * csA];
        af.y = A[aRow + (long)(ka + 1) * csA];
        bf.x = Bm[(long)(k0 + half)     * rsB + bCol];
        bf.y = Bm[(long)(k0 + 2 + half) * rsB + bCol];
        acc = __builtin_amdgcn_wmma_f32_16x16x4_f32(
                  false, af, false, bf, (short)0, acc, false, false);
    }

    const int n  = tn + l15;
    const int mb = tm + (half << 3);
    float badd = 0.f;
    if (bias) badd = bias[(long)bo * sBias + n];
#pragma unroll
    for (int i = 0; i < 8; ++i) {
        float v = acc[i] * scale + badd;
        if (act == 1) v = siluf(v);
        C[(long)(mb + i) * rsC + n] = v;
    }
}

// ---------------------------------------------------------------------------
__global__ void k_embed(const int* __restrict__ x, const float* __restrict__ emb,
                        float* __restrict__ out)
{
    const int tok = blockIdx.x;
    const long id = (long)x[tok];
    const float* src = emb + id * DIMV;
    float* dst = out + (long)tok * DIMV;
    for (int j = threadIdx.x; j < DIMV; j += blockDim.x) dst[j] = src[j];
}

__global__ void k_init(const float* __restrict__ mW1, const float* __restrict__ mb1,
                       const float* __restrict__ mW2, const float* __restrict__ mb2,
                       float* __restrict__ p, float* __restrict__ m)
{
    const long i = (long)blockIdx.x * blockDim.x + threadIdx.x;
    if (i >= (long)BATCH * PBLOB) return;
    const long off = i % PBLOB;
    float v;
    if      (off < OFF_B1) v = mW1[off];
    else if (off < OFF_W2) v = mb1[off - OFF_B1];
    else if (off < OFF_B2) v = mW2[off - OFF_W2];
    else                   v = mb2[off - OFF_B2];
    p[i] = v;
    m[i] = 0.f;
}

__global__ void k_l2n(const float* __restrict__ in, float* __restrict__ out)
{
    const long row = blockIdx.x;
    const float* r = in  + row * DIMV;
    float*       w = out + row * DIMV;
    __shared__ float red[256];
    const int tid = threadIdx.x;
    float s = 0.f;
    for (int j = tid; j < DIMV; j += 256) { float v = r[j]; s += v * v; }
    red[tid] = s; __syncthreads();
    for (int st = 128; st > 0; st >>= 1) {
        if (tid < st) red[tid] += red[tid + st];
        __syncthreads();
    }
    const float inv = 1.f / (sqrtf(red[0]) + 1e-6f);
    for (int j = tid; j < DIMV; j += 256) w[j] = r[j] * inv;
}

__global__ void k_concat(const float* __restrict__ persistent,
                         const float* __restrict__ h,
                         const float* __restrict__ seg, long segBatchStride,
                         float* __restrict__ comb)
{
    const int z = blockIdx.x;
    const int b = z / T_CTX, t = z - b * T_CTX;
    const float* src;
    if (t < NPT)            src = persistent + (long)t * DIMV;
    else if (t < NPT+SEGL)  src = h   + ((long)b * SEGL + (t - NPT)) * DIMV;
    else                    src = seg + (long)b * segBatchStride + (long)(t - NPT - SEGL) * DIMV;
    float* dst = comb + (long)z * DIMV;
    for (int j = threadIdx.x; j < DIMV; j += blockDim.x) dst[j] = src[j];
}

__global__ void k_softmax_causal(float* __restrict__ scores)
{
    const int z = blockIdx.x;
    const int q = NPT + SEGL + (z % SEGL);
    float* row = scores + (long)z * T_CTX;
    __shared__ float red[128];
    const int tid = threadIdx.x;

    float mx = -INFINITY;
    for (int j = tid; j < T_CTX; j += 128) {
        float v = (j > q) ? -INFINITY : row[j];
        mx = fmaxf(mx, v);
    }
    red[tid] = mx; __syncthreads();
    for (int s = 64; s > 0; s >>= 1) {
        if (tid < s) red[tid] = fmaxf(red[tid], red[tid + s]);
        __syncthreads();
    }
    mx = red[0]; __syncthreads();

    float sum = 0.f;
    for (int j = tid; j < T_CTX; j += 128) {
        float v = (j > q) ? 0.f : __expf(row[j] - mx);
        row[j] = v;
        sum += v;
    }
    red[tid] = sum; __syncthreads();
    for (int s = 64; s > 0; s >>= 1) {
        if (tid < s) red[tid] += red[tid + s];
        __syncthreads();
    }
    const float inv = 1.f / red[0];
    for (int j = tid; j < T_CTX; j += 128) row[j] *= inv;
}

// ---------------------------------------------------------------------------
// Sequential test-time memory update (one WG per sample, 128 tokens).
// ---------------------------------------------------------------------------
__global__ void __launch_bounds__(1024)
k_tokloop(const float* __restrict__ Kbuf, const float* __restrict__ Vbuf,
          float* __restrict__ p, float* __restrict__ m)
{
    const int b = blockIdx.x;
    float* W1  = p + (long)b * PBLOB + OFF_W1;
    float* b1  = p + (long)b * PBLOB + OFF_B1;
    float* W2  = p + (long)b * PBLOB + OFF_W2;
    float* b2  = p + (long)b * PBLOB + OFF_B2;
    float* mW1 = m + (long)b * PBLOB + OFF_W1;
    float* mb1 = m + (long)b * PBLOB + OFF_B1;
    float* mW2 = m + (long)b * PBLOB + OFF_W2;
    float* mb2 = m + (long)b * PBLOB + OFF_B2;
    const float* Kb = Kbuf + (long)b * SEGL * DIMV;
    const float* Vb = Vbuf + (long)b * SEGL * DIMV;

    __shared__ float k_s[DIMV];
    __shared__ float e_s[DIMV];
    __shared__ float h1_s[WIDTHM];
    __shared__ float pre_s[WIDTHM];
    __shared__ float dpre_s[WIDTHM];
    __shared__ float part[8][WIDTHM];

    const int tid = threadIdx.x;
    const int jg  = tid & (WIDTHM - 1);
    const int ig  = tid >> 7;

    for (int t = 0; t < SEGL; ++t) {
        const float* kv = Kb + (long)t * DIMV;
        const float* vv = Vb + (long)t * DIMV;
        k_s[tid] = kv[tid];
        __syncthreads();

        float s = 0.f;
        for (int i = ig * 128; i < ig * 128 + 128; ++i)
            s += k_s[i] * W1[(long)i * WIDTHM + jg];
        part[ig][jg] = s;
        __syncthreads();
        if (tid < WIDTHM) {
            float acc = b1[tid];
#pragma unroll
            for (int r = 0; r < 8; ++r) acc += part[r][tid];
            pre_s[tid] = acc;
            h1_s[tid]  = siluf(acc);
        }
        __syncthreads();

        {
            float acc = b2[tid];
            for (int j = 0; j < WIDTHM; ++j)
                acc += h1_s[j] * W2[(long)j * DIMV + tid];
            e_s[tid] = 2.f * (acc - vv[tid]);
        }
        __syncthreads();

        s = 0.f;
        for (int d = ig * 128; d < ig * 128 + 128; ++d)
            s += e_s[d] * W2[(long)jg * DIMV + d];
        part[ig][jg] = s;
        __syncthreads();
        if (tid < WIDTHM) {
            float dh1 = 0.f;
#pragma unroll
            for (int r = 0; r < 8; ++r) dh1 += part[r][tid];
            const float x  = pre_s[tid];
            const float sg = 1.f / (1.f + __expf(-x));
            dpre_s[tid] = dh1 * sg * (1.f + x * (1.f - sg));
        }
        __syncthreads();

        for (int c = 0; c < WIDTHM; ++c) {
            const long idx = (long)c * 1024 + tid;
            const int  i = (int)(idx >> 7), j = (int)(idx & 127);
            const float g  = k_s[i] * dpre_s[j];
            const float mm = 0.9f * mW1[idx] - 0.1f * g;
            mW1[idx] = mm;
            W1[idx]  = 0.99f * W1[idx] + mm;
        }
        for (int c = 0; c < WIDTHM; ++c) {
            const long idx = (long)c * 1024 + tid;
            const float g  = h1_s[c] * e_s[tid];
            const float mm = 0.9f * mW2[idx] - 0.1f * g;
            mW2[idx] = mm;
            W2[idx]  = 0.99f * W2[idx] + mm;
        }
        {
            const float g  = e_s[tid];
            const float mm = 0.9f * mb2[tid] - 0.1f * g;
            mb2[tid] = mm;
            b2[tid]  = 0.99f * b2[tid] + mm;
        }
        if (tid < WIDTHM) {
            const float g  = dpre_s[tid];
            const float mm = 0.9f * mb1[tid] - 0.1f * g;
            mb1[tid] = mm;
            b1[tid]  = 0.99f * b1[tid] + mm;
        }
        __threadfence_block();
        __syncthreads();
    }
}

__global__ void k_mul(const float* __restrict__ a, const float* __restrict__ bm,
                      float* __restrict__ out, long outBatchStride)
{
    const int z = blockIdx.x;
    const int b = z / SEGL, t = z - b * SEGL;
    const float* pa = a  + (long)z * DIMV;
    const float* pb = bm + (long)z * DIMV;
    float* po = out + (long)b * outBatchStride + (long)t * DIMV;
    for (int j = threadIdx.x; j < DIMV; j += blockDim.x) po[j] = pa[j] * pb[j];
}

// ---------------------------------------------------------------------------
extern "C" void kernel_launch(void* const* d_in, const int* in_sizes, int n_in,
                              void* d_out, int out_size, void* d_ws, size_t ws_size,
                              hipStream_t stream)
{
    const int*   x          = (const int*)  d_in[0];
    const float* embedding  = (const float*)d_in[1];
    const float* persistent = (const float*)d_in[2];
    const float* Wq = (const float*)d_in[3];  const float* bq = (const float*)d_in[4];
    const float* Wk = (const float*)d_in[5];  const float* bk = (const float*)d_in[6];
    const float* Wv = (const float*)d_in[7];  const float* bv = (const float*)d_in[8];
    const float* attn_in_w  = (const float*)d_in[9];
    const float* attn_in_b  = (const float*)d_in[10];
    const float* attn_out_w = (const float*)d_in[11];
    const float* attn_out_b = (const float*)d_in[12];
    const float* mW1 = (const float*)d_in[13]; const float* mb1 = (const float*)d_in[14];
    const float* mW2 = (const float*)d_in[15]; const float* mb2 = (const float*)d_in[16];
    const float* head_w = (const float*)d_in[17];
    const float* head_b = (const float*)d_in[18];
    float* out = (float*)d_out;

    float* ws = (float*)d_ws;
    size_t o = 0;
    auto alloc = [&](size_t n) { float* q = ws + o; o += n; return q; };
    float* x_emb  = alloc((size_t)BATCH * NTOK * DIMV);
    float* pblob  = alloc((size_t)BATCH * PBLOB);
    float* mblob  = alloc((size_t)BATCH * PBLOB);
    float* qk     = alloc((size_t)BATCH * SEGL * DIMV);
    float* h1     = alloc((size_t)BATCH * SEGL * WIDTHM);
    float* hbuf   = alloc((size_t)BATCH * SEGL * DIMV);
    float* comb   = alloc((size_t)BATCH * T_CTX * DIMV);
    float* qkv    = alloc((size_t)BATCH * T_CTX * 3 * DIMV);
    float* scores = alloc((size_t)BATCH * HEADS * SEGL * T_CTX);
    float* ctx    = alloc((size_t)BATCH * SEGL * DIMV);
    float* segout = alloc((size_t)BATCH * SEGL * DIMV);
    float* kbuf   = alloc((size_t)BATCH * SEGL * DIMV);
    float* vbuf   = alloc((size_t)BATCH * SEGL * DIMV);
    float* qo     = alloc((size_t)BATCH * SEGL * DIMV);
    float* memout = alloc((size_t)BATCH * SEGL * DIMV);
    float* full   = alloc((size_t)BATCH * NTOK * DIMV);
    (void)ws_size; (void)in_sizes; (void)n_in; (void)out_size;

    auto gemmT = [&](const float* A, const float* Bm, const float* bias, float* C,
                     int M, int N, int K, long rsA, long rsB, long rsC,
                     int nbatch, long sA, long sB, long sC, long sBias,
                     float scale, int act) {
        dim3 g(N / TN, (M + TM - 1) / TM, nbatch);
        k_gemm_tiled<<<g, dim3(256), 0, stream>>>(A, Bm, bias, C, M, N, K,
                                                  rsA, rsB, rsC, sA, sB, sC,
                                                  sBias, scale, act);
    };
    auto gemmG = [&](const float* A, const float* Bm, const float* bias, float* C,
                     int M, int N, int K,
                     long rsA, long csA, long rsB, long csB, long rsC,
                     int bOuter, int bInner,
                     long sAo, long sAi, long sBo, long sBi, long sCo, long sCi,
                     long sBias, float scale, int act) {
        dim3 g(N / 16, M / 16, bOuter * bInner);
        k_gemm<<<g, dim3(32), 0, stream>>>(A, Bm, bias, C, M, N, K,
                                           rsA, csA, rsB, csB, rsC, bInner,
                                           sAo, sAi, sBo, sBi, sCo, sCi, sBias,
                                           scale, act);
    };

    k_embed<<<BATCH * NTOK, 256, 0, stream>>>(x, embedding, x_emb);
    {
        const long tot = (long)BATCH * PBLOB;
        k_init<<<(unsigned)((tot + 255) / 256), 256, 0, stream>>>(mW1, mb1, mW2, mb2,
                                                                  pblob, mblob);
    }

    const long sTok = (long)NTOK * DIMV;

    for (int s = 0; s < NSEG; ++s) {
        const float* seg = x_emb + (long)s * SEGL * DIMV;

        // qk = l2n(seg @ Wq + bq)
        gemmT(seg, Wq, bq, qk, SEGL, DIMV, DIMV, DIMV, DIMV, DIMV,
              BATCH, sTok, 0, (long)SEGL * DIMV, 0, 1.f, 0);
        k_l2n<<<BATCH * SEGL, 256, 0, stream>>>(qk, qk);

        // h = mlp(p, qk)  (per-sample weights)
        gemmT(qk, pblob + OFF_W1, pblob + OFF_B1, h1, SEGL, WIDTHM, DIMV,
              DIMV, WIDTHM, WIDTHM,
              BATCH, (long)SEGL * DIMV, PBLOB, (long)SEGL * WIDTHM, PBLOB, 1.f, 1);
        gemmT(h1, pblob + OFF_W2, pblob + OFF_B2, hbuf, SEGL, DIMV, WIDTHM,
              WIDTHM, DIMV, DIMV,
              BATCH, (long)SEGL * WIDTHM, PBLOB, (long)SEGL * DIMV, PBLOB, 1.f, 0);

        // combined = [persistent | h | seg]
        k_concat<<<BATCH * T_CTX, 256, 0, stream>>>(persistent, hbuf, seg, sTok, comb);

        // qkv = combined @ attn_in_w + attn_in_b   (M=272 ragged)
        gemmT(comb, attn_in_w, attn_in_b, qkv, T_CTX, 3 * DIMV, DIMV,
              DIMV, 3L * DIMV, 3L * DIMV,
              BATCH, (long)T_CTX * DIMV, 0, (long)T_CTX * 3 * DIMV, 0, 1.f, 0);

        // scores = Q[144..271] . K^T / 8
        gemmG(qkv + (long)(NPT + SEGL) * 3 * DIMV, qkv + DIMV, nullptr, scores,
              SEGL, T_CTX, HD,
              3L * DIMV, 1, 1, 3L * DIMV, T_CTX,
              BATCH, HEADS,
              (long)T_CTX * 3 * DIMV, HD, (long)T_CTX * 3 * DIMV, HD,
              (long)HEADS * SEGL * T_CTX, (long)SEGL * T_CTX,
              0, 0.125f, 0);

        k_softmax_causal<<<BATCH * HEADS * SEGL, 128, 0, stream>>>(scores);

        // ctx = softmax @ V
        gemmG(scores, qkv + 2L * DIMV, nullptr, ctx,
              SEGL, HD, T_CTX,
              T_CTX, 1, 3L * DIMV, 1, DIMV,
              BATCH, HEADS,
              (long)HEADS * SEGL * T_CTX, (long)SEGL * T_CTX,
              (long)T_CTX * 3 * DIMV, HD,
              (long)SEGL * DIMV, HD,
              0, 1.f, 0);

        // seg_out = ctx @ attn_out_w + attn_out_b
        gemmT(ctx, attn_out_w, attn_out_b, segout, SEGL, DIMV, DIMV,
              DIMV, DIMV, DIMV,
              BATCH, (long)SEGL * DIMV, 0, (long)SEGL * DIMV, 0, 1.f, 0);

        // K = l2n(seg_out @ Wk + bk); V = seg_out @ Wv + bv
        gemmT(segout, Wk, bk, kbuf, SEGL, DIMV, DIMV, DIMV, DIMV, DIMV,
              BATCH, (long)SEGL * DIMV, 0, (long)SEGL * DIMV, 0, 1.f, 0);
        k_l2n<<<BATCH * SEGL, 256, 0, stream>>>(kbuf, kbuf);
        gemmT(segout, Wv, bv, vbuf, SEGL, DIMV, DIMV, DIMV, DIMV, DIMV,
              BATCH, (long)SEGL * DIMV, 0, (long)SEGL * DIMV, 0, 1.f, 0);

        // 128 sequential test-time gradient steps
        k_tokloop<<<BATCH, 1024, 0, stream>>>(kbuf, vbuf, pblob, mblob);

        // mem_out = mlp(p, l2n(seg_out @ Wq + bq))
        gemmT(segout, Wq, bq, qo, SEGL, DIMV, DIMV, DIMV, DIMV, DIMV,
              BATCH, (long)SEGL * DIMV, 0, (long)SEGL * DIMV, 0, 1.f, 0);
        k_l2n<<<BATCH * SEGL, 256, 0, stream>>>(qo, qo);
        gemmT(qo, pblob + OFF_W1, pblob + OFF_B1, h1, SEGL, WIDTHM, DIMV,
              DIMV, WIDTHM, WIDTHM,
              BATCH, (long)SEGL * DIMV, PBLOB, (long)SEGL * WIDTHM, PBLOB, 1.f, 1);
        gemmT(h1, pblob + OFF_W2, pblob + OFF_B2, memout, SEGL, DIMV, WIDTHM,
              WIDTHM, DIMV, DIMV,
              BATCH, (long)SEGL * WIDTHM, PBLOB, (long)SEGL * DIMV, PBLOB, 1.f, 0);

        k_mul<<<BATCH * SEGL, 256, 0, stream>>>(segout, memout,
                                                full + (long)s * SEGL * DIMV, sTok);
    }

    // head: logits = full @ head_w + head_b
    gemmT(full, head_w, head_b, out, BATCH * NTOK, VOCAB, DIMV,
          DIMV, VOCAB, VOCAB, 1, 0, 0, 0, 0, 1.f, 0);
}